// CorrLayer_58961311040340
// MI455X (gfx1250) — compile-verified
//
#include <hip/hip_runtime.h>

// FlowNet-style local correlation as a banded fp32 GEMM on CDNA5 WMMA.
// out[b, oi*9+oj, h, w] = (1/C) * sum_c x[b,c,h,w] * y_pad[b,c,h+oi,w+oj]
//
// Per (b, h, oi): G[w, w'] = sum_c x[c,w] * y[c, h+oi-4, w'] ; outputs are the
// band w'-w in [-4, 4]. Each wave owns a 16-wide w tile and computes two
// 16x16 fp32 WMMA tiles of G (cols w0-4..w0+11 and w0+12..w0+27), K = C = 128
// chained through 32 x V_WMMA_F32_16X16X4_F32 per tile (split into 4
// independent accumulator chains for ILP), then scatters the in-band elements.
// Out-of-range y columns are loaded from clamped addresses (garbage only
// contaminates their own product column) and zeroed at store time.

typedef __attribute__((ext_vector_type(2))) float v2f;
typedef __attribute__((ext_vector_type(8))) float v8f;

#define C_DIM 128
#define H_DIM 192
#define W_DIM 192
#define B_DIM 4
#define KSZ   9
#define RAD   4
#define HW    (H_DIM * W_DIM)          // 36864
#define CHW   ((size_t)C_DIM * HW)     // 4718592

__launch_bounds__(32)
__global__ void corr_wmma_f32(const float* __restrict__ x,
                              const float* __restrict__ y,
                              float* __restrict__ out)
{
    const int tw = blockIdx.x;          // w tile index (0..11)
    const int h  = blockIdx.y;          // 0..191
    const int b  = blockIdx.z;          // 0..3
    const int w0 = tw * 16;

    const int lane = threadIdx.x;       // 0..31 (wave32)
    const int half = lane >> 4;         // 0 or 1
    const int lm   = lane & 15;         // M row (A) / N col (B) / col n (D)

    // ---------------- A tile: x[16 pixels x 128 channels] into registers ---
    // WMMA f32 16x16x4 A layout: lane L holds M = L%16; VGPR0 = K(2*half),
    // VGPR1 = K(2*half + 1) within each K=4 chunk.
    const float* xbase = x + (size_t)b * CHW + (size_t)(2 * half) * HW
                           + (size_t)h * W_DIM + (w0 + lm);
    v2f a[32];
#pragma unroll
    for (int kc = 0; kc < 32; ++kc) {
        const float* p = xbase + (size_t)(4 * kc) * HW;
        a[kc].x = p[0];
        a[kc].y = p[HW];
    }

    // ---------------- per-lane B column positions + validity ---------------
    const int wp0 = w0 - 4 + lm;        // tile0 column w'
    const int wp1 = w0 + 12 + lm;       // tile1 column w'
    const bool ok0 = (wp0 >= 0) && (wp0 < W_DIM);
    const bool ok1 = (wp1 < W_DIM);     // wp1 >= 12 always
    const int wp0c = wp0 < 0 ? 0 : (wp0 >= W_DIM ? W_DIM - 1 : wp0);
    const int wp1c = wp1 >= W_DIM ? W_DIM - 1 : wp1;

    const size_t ych = (size_t)b * CHW + (size_t)(2 * half) * HW;
    const float inv_c = 1.0f / (float)C_DIM;

    for (int oi = 0; oi < KSZ; ++oi) {
        v8f d0a = {}, d0b = {};         // tile0: even / odd K-chunk chains
        v8f d1a = {}, d1b = {};         // tile1: even / odd K-chunk chains
        const int row = h + oi - RAD;
        if (row >= 0 && row < H_DIM) {
            const float* y0 = y + ych + (size_t)row * W_DIM + wp0c;
            const float* y1 = y + ych + (size_t)row * W_DIM + wp1c;
#pragma unroll
            for (int kc = 0; kc < 32; kc += 2) {
                const size_t offA = (size_t)(4 * kc) * HW;
                const size_t offB = (size_t)(4 * (kc + 1)) * HW;
                v2f b0a, b1a, b0b, b1b;
                b0a.x = y0[offA];
                b0a.y = y0[offA + HW];
                b1a.x = y1[offA];
                b1a.y = y1[offA + HW];
                b0b.x = y0[offB];
                b0b.y = y0[offB + HW];
                b1b.x = y1[offB];
                b1b.y = y1[offB + HW];
                d0a = __builtin_amdgcn_wmma_f32_16x16x4_f32(
                        false, a[kc],     false, b0a, (short)0, d0a, false, false);
                d1a = __builtin_amdgcn_wmma_f32_16x16x4_f32(
                        false, a[kc],     false, b1a, (short)0, d1a, false, false);
                d0b = __builtin_amdgcn_wmma_f32_16x16x4_f32(
                        false, a[kc + 1], false, b0b, (short)0, d0b, false, false);
                d1b = __builtin_amdgcn_wmma_f32_16x16x4_f32(
                        false, a[kc + 1], false, b1b, (short)0, d1b, false, false);
            }
        }
        // ------------- band extraction + store (also writes zeros) ---------
        // D layout: lane L holds G[v + 8*half][lm] in d[v].
        float* obase = out + (size_t)b * (KSZ * KSZ) * HW
                           + (size_t)(oi * KSZ) * HW
                           + (size_t)h * W_DIM + w0;
#pragma unroll
        for (int v = 0; v < 8; ++v) {
            const int m = v + 8 * half;     // output pixel within tile
            const int oj0 = lm - m;         // tile0: col w0-4+lm = w0+m+oj-4
            if (oj0 >= 0 && oj0 <= 8) {
                const float r = ok0 ? (d0a[v] + d0b[v]) * inv_c : 0.0f;
                obase[(size_t)oj0 * HW + m] = r;
            }
            const int oj1 = lm + 16 - m;    // tile1: col w0+12+lm
            if (oj1 <= 8) {                 // >= 0 always
                const float r = ok1 ? (d1a[v] + d1b[v]) * inv_c : 0.0f;
                obase[(size_t)oj1 * HW + m] = r;
            }
        }
    }
}

extern "C" void kernel_launch(void* const* d_in, const int* in_sizes, int n_in,
                              void* d_out, int out_size, void* d_ws, size_t ws_size,
                              hipStream_t stream) {
    (void)in_sizes; (void)n_in; (void)d_ws; (void)ws_size; (void)out_size;
    const float* x = (const float*)d_in[0];
    const float* y = (const float*)d_in[1];
    float* out = (float*)d_out;
    dim3 grid(W_DIM / 16, H_DIM, B_DIM);
    dim3 block(32, 1, 1);
    corr_wmma_f32<<<grid, block, 0, stream>>>(x, y, out);
}